// COSLoss_52905407152941
// MI455X (gfx1250) — compile-verified
//
#include <hip/hip_runtime.h>

// ---------------------------------------------------------------------------
// Problem constants (fixed by the reference setup):
//   pd, gt      : (8, 4, 256, 256) f32
//   pd/gt masks : (8, 32, 256, 256) i32 in {0,1}
// Memory-bound: 144.7 MB single-pass => ~6.2 us floor at 23.3 TB/s.
// ---------------------------------------------------------------------------

typedef __attribute__((ext_vector_type(2))) float v2f;
typedef __attribute__((ext_vector_type(8))) float v8f;

// Full wave32 sum using two chained V_WMMA_F32_16X16X4_F32 with an all-ones
// B matrix. With B == ones, D[m,n] = sum_k A[m,k]; each lane's partial is
// placed exactly once in A (other A VGPR = 0), so:
//   stage 1: lane L's 8 D VGPRs hold rowsums; their sum s is
//            sum(partials of lanes 0..7 and 16..23) in lanes 0..15 and
//            sum(partials of lanes 8..15 and 24..31) in lanes 16..31.
//   stage 2: same trick on s => every lane's D[0] = total wave sum.
// Row-sums over K are invariant to any K/lane permutation, so this is
// layout-robust. EXEC is all-ones at every call site (no divergence).
__device__ __forceinline__ float wave_reduce32(float x) {
#if __has_builtin(__builtin_amdgcn_wmma_f32_16x16x4_f32)
  v2f a;    a[0] = x;    a[1] = 0.0f;
  v2f ones; ones[0] = 1.0f; ones[1] = 1.0f;
  v8f c = {};
  v8f d1 = __builtin_amdgcn_wmma_f32_16x16x4_f32(
      false, a, false, ones, (short)0, c, false, false);
  float s = ((d1[0] + d1[1]) + (d1[2] + d1[3])) +
            ((d1[4] + d1[5]) + (d1[6] + d1[7]));
  v2f a2; a2[0] = s; a2[1] = 0.0f;
  v8f d2 = __builtin_amdgcn_wmma_f32_16x16x4_f32(
      false, a2, false, ones, (short)0, c, false, false);
  return d2[0];
#else
  // Fallback (should not trigger on gfx1250): shuffle tree.
  for (int off = 16; off > 0; off >>= 1) x += __shfl_xor(x, off, 32);
  return x;
#endif
}

// Per-pixel contribution:
//   pd3^2*Sp + gt3^2*Sg - (dot3 + 2*pd3*gt3)*Sj
__device__ __forceinline__ float pix_term(float p0, float p1, float p2, float p3,
                                          float g0, float g1, float g2, float g3,
                                          int sp, int sg, int sj) {
  float d3 = fmaf(p0, g0, fmaf(p1, g1, p2 * g2));
  return fmaf(p3 * p3, (float)sp,
         fmaf(g3 * g3, (float)sg,
              -(d3 + 2.0f * p3 * g3) * (float)sj));
}

// 512 blocks x 256 threads, 4 pixels/thread: 524288 pixels exactly.
__global__ __launch_bounds__(256) void cos_loss_main(
    const float* __restrict__ pd, const float* __restrict__ gt,
    const int* __restrict__ pdm, const int* __restrict__ gtm,
    float* __restrict__ blockPartials) {
  const int tid = blockIdx.x * 256 + threadIdx.x;
  const int p0i = tid << 2;          // first flat pixel index (b*65536 + hw)
  const int b   = p0i >> 16;
  const int hw  = p0i & 0xFFFF;      // multiple of 4 -> 16B aligned accesses

  // pd/gt channel c lives at element offset (b*4 + c)*65536 + hw.
  const int cb = (b << 18) + hw;
  const float4 pA = *reinterpret_cast<const float4*>(pd + cb);
  const float4 pB = *reinterpret_cast<const float4*>(pd + cb + (1 << 16));
  const float4 pC = *reinterpret_cast<const float4*>(pd + cb + (2 << 16));
  const float4 pD = *reinterpret_cast<const float4*>(pd + cb + (3 << 16));
  const float4 gA = *reinterpret_cast<const float4*>(gt + cb);
  const float4 gB = *reinterpret_cast<const float4*>(gt + cb + (1 << 16));
  const float4 gC = *reinterpret_cast<const float4*>(gt + cb + (2 << 16));
  const float4 gD = *reinterpret_cast<const float4*>(gt + cb + (3 << 16));

  // Mask reductions over n=0..31. Adjacent lanes read adjacent int4s ->
  // fully coalesced 512B/wave B128 loads; 64 independent loads/thread.
  int4 Sp = make_int4(0, 0, 0, 0);
  int4 Sg = make_int4(0, 0, 0, 0);
  int4 Sj = make_int4(0, 0, 0, 0);
  const int mb = (b << 21) + hw;     // (b*32)*65536 + hw
#pragma unroll 8
  for (int n = 0; n < 32; ++n) {
    const int4 pm4 = *reinterpret_cast<const int4*>(pdm + mb + (n << 16));
    const int4 gm4 = *reinterpret_cast<const int4*>(gtm + mb + (n << 16));
    Sp.x += pm4.x; Sp.y += pm4.y; Sp.z += pm4.z; Sp.w += pm4.w;
    Sg.x += gm4.x; Sg.y += gm4.y; Sg.z += gm4.z; Sg.w += gm4.w;
    Sj.x += pm4.x & gm4.x; Sj.y += pm4.y & gm4.y;   // 0/1 masks: * == &
    Sj.z += pm4.z & gm4.z; Sj.w += pm4.w & gm4.w;
  }

  float acc =
      pix_term(pA.x, pB.x, pC.x, pD.x, gA.x, gB.x, gC.x, gD.x, Sp.x, Sg.x, Sj.x) +
      pix_term(pA.y, pB.y, pC.y, pD.y, gA.y, gB.y, gC.y, gD.y, Sp.y, Sg.y, Sj.y) +
      pix_term(pA.z, pB.z, pC.z, pD.z, gA.z, gB.z, gC.z, gD.z, Sp.z, Sg.z, Sj.z) +
      pix_term(pA.w, pB.w, pC.w, pD.w, gA.w, gB.w, gC.w, gD.w, Sp.w, Sg.w, Sj.w);

  // Wave reduce via WMMA, then deterministic block reduce via LDS.
  const float wr = wave_reduce32(acc);
  __shared__ float wsum[8];
  const int lane = threadIdx.x & 31;
  const int wid  = threadIdx.x >> 5;
  if (lane == 0) wsum[wid] = wr;
  __syncthreads();
  if (threadIdx.x == 0) {
    float s = 0.0f;
#pragma unroll
    for (int i = 0; i < 8; ++i) s += wsum[i];
    blockPartials[blockIdx.x] = s;
  }
}

// Single block of 512 threads (16 waves) folds the 512 block partials and
// applies the final 1 + sum/N. Deterministic tree, no atomics.
__global__ __launch_bounds__(512) void cos_loss_final(
    const float* __restrict__ blockPartials, float* __restrict__ out) {
  const float x  = blockPartials[threadIdx.x];
  const float wr = wave_reduce32(x);
  __shared__ float wsum[16];
  const int lane = threadIdx.x & 31;
  const int wid  = threadIdx.x >> 5;
  if (lane == 0) wsum[wid] = wr;
  __syncthreads();
  if (threadIdx.x == 0) {
    float s = 0.0f;
#pragma unroll
    for (int i = 0; i < 16; ++i) s += wsum[i];
    out[0] = 1.0f + s * (1.0f / 16777216.0f);   // N = 8*32*256*256
  }
}

extern "C" void kernel_launch(void* const* d_in, const int* in_sizes, int n_in,
                              void* d_out, int out_size, void* d_ws, size_t ws_size,
                              hipStream_t stream) {
  (void)in_sizes; (void)n_in; (void)out_size; (void)ws_size;
  const float* pd  = (const float*)d_in[0];
  const float* gt  = (const float*)d_in[1];
  const int*   pdm = (const int*)d_in[2];
  const int*   gtm = (const int*)d_in[3];
  float* partials  = (float*)d_ws;              // 512 floats of scratch
  float* out       = (float*)d_out;

  cos_loss_main<<<512, 256, 0, stream>>>(pd, gt, pdm, gtm, partials);
  cos_loss_final<<<1, 512, 0, stream>>>(partials, out);
}